// EdgeEmbedding_36644660969761
// MI455X (gfx1250) — compile-verified
//
#include <hip/hip_runtime.h>

typedef __attribute__((ext_vector_type(2))) float v2f;
typedef __attribute__((ext_vector_type(8))) float v8f;

#define ZROW 36
#define C_EMB 32
#define N_RBF 32
#define ZMAX 119

__global__ __launch_bounds__(256) void edge_embed_kernel(
    const int* __restrict__ an, const int* __restrict__ nbr,
    const float* __restrict__ ev, const float* __restrict__ zt,
    const float* __restrict__ zw, const float* __restrict__ dw,
    const float* __restrict__ db, const float* __restrict__ rbeta,
    const float* __restrict__ rcen, float* __restrict__ out,
    int nEdges, int nTiles)
{
  __shared__ float  ldsZ[ZMAX * ZROW];
  __shared__ float  ldsBz[16 * 2 * 32 * 2];
  __shared__ float  ldsBd[8 * 6 * 32 * 2];
  __shared__ float  ldsCen[N_RBF];
  __shared__ float  ldsBeta[N_RBF];
  __shared__ float4 ldsG[8 * 16];

  const int tid = threadIdx.x;

  for (int i = tid; i < ZMAX * C_EMB; i += blockDim.x) {
    int row = i >> 5, col = i & 31;
    ldsZ[row * ZROW + col] = zt[i];
  }
  for (int i = tid; i < 16 * 2 * 32 * 2; i += blockDim.x) {
    int j = i & 1, ln = (i >> 1) & 31, nt = (i >> 6) & 1, kt = i >> 7;
    int k = kt * 4 + ((ln & 16) >> 3) + j;
    int n = nt * 16 + (ln & 15);
    ldsBz[i] = zw[n * 64 + k];
  }
  for (int i = tid; i < 8 * 6 * 32 * 2; i += blockDim.x) {
    int j = i & 1, ln = (i >> 1) & 31;
    int g = i >> 6;
    int nt = g % 6, kt = g / 6;
    int k = kt * 4 + ((ln & 16) >> 3) + j;
    int m = nt * 16 + (ln & 15);
    ldsBd[i] = dw[m * 32 + k];
  }
  if (tid < N_RBF) { ldsCen[tid] = rcen[tid]; ldsBeta[tid] = rbeta[tid]; }
  __syncthreads();

  const int wave = tid >> 5;
  const int lane = tid & 31;
  const int tileIdx = blockIdx.x * (blockDim.x >> 5) + wave;
  if (tileIdx >= nTiles) return;

  const int e0 = tileIdx << 4;
  const int le = lane & 15;
  int myE = e0 + le;
  if (myE >= nEdges) myE = nEdges - 1;
  const int koff = (lane & 16) >> 3;

  const int ii = nbr[myE];
  const int jj = nbr[nEdges + myE];
  const int zi = an[ii] * ZROW;
  const int zj = an[jj] * ZROW;

  float vx = ev[myE * 3 + 0], vy = ev[myE * 3 + 1], vz = ev[myE * 3 + 2];
  const float r = sqrtf(vx * vx + vy * vy + vz * vz);
  const float inv = 1.0f / r;
  vx *= inv; vy *= inv; vz *= inv;
  const float env = (r < 5.0f) ? 0.5f * (__cosf(r * 0.62831853f) + 1.0f) : 0.0f;
  const float texp = __expf(-r);

  if (lane < 16) ldsG[wave * 16 + le] = make_float4(vx, vy, vz, env);

  float bias[6];
  #pragma unroll
  for (int t = 0; t < 6; ++t) bias[t] = db[t * 16 + le];

  const v8f z8 = {0.f, 0.f, 0.f, 0.f, 0.f, 0.f, 0.f, 0.f};

  v8f accz[2] = { z8, z8 };
  #pragma unroll
  for (int s = 0; s < 16; ++s) {
    const int k = 4 * s + koff;
    v2f a;
    if (s < 8) a = *(const v2f*)&ldsZ[zi + k];
    else       a = *(const v2f*)&ldsZ[zj + (k - 32)];
    #pragma unroll
    for (int nt = 0; nt < 2; ++nt) {
      v2f b = *(const v2f*)&ldsBz[(s * 2 + nt) * 64 + lane * 2];
      accz[nt] = __builtin_amdgcn_wmma_f32_16x16x4_f32(
          false, a, false, b, (short)0, accz[nt], false, false);
    }
  }

  v8f accr[6] = { z8, z8, z8, z8, z8, z8 };
  #pragma unroll
  for (int s = 0; s < 8; ++s) {
    const int k = 4 * s + koff;
    const float d0 = texp - ldsCen[k];
    const float d1 = texp - ldsCen[k + 1];
    v2f a;
    a.x = __expf(-ldsBeta[k]     * d0 * d0);
    a.y = __expf(-ldsBeta[k + 1] * d1 * d1);
    #pragma unroll
    for (int nt = 0; nt < 6; ++nt) {
      v2f b = *(const v2f*)&ldsBd[(s * 6 + nt) * 64 + lane * 2];
      accr[nt] = __builtin_amdgcn_wmma_f32_16x16x4_f32(
          false, a, false, b, (short)0, accr[nt], false, false);
    }
  }

  const long outStride = (long)nEdges * 288;
  float* const outI = out;
  float* const outA = out + outStride;
  float* const outS = out + 2 * outStride;
  const int hbase = (lane & 16) >> 1;

  #pragma unroll
  for (int v = 0; v < 8; ++v) {
    const int el = hbase + v;
    const int eg = e0 + el;
    if (eg >= nEdges) continue;
    const float4 g = ldsG[wave * 16 + el];
    const float tm  = (g.x * g.x + g.y * g.y + g.z * g.z) * (1.0f / 3.0f);
    const float s00 = g.x * g.x - tm, s11 = g.y * g.y - tm, s22 = g.z * g.z - tm;
    const float s01 = g.x * g.y, s02 = g.x * g.z, s12 = g.y * g.z;
    const long rowbase = (long)eg * 288 + (long)le * 9;
    #pragma unroll
    for (int t = 0; t < 2; ++t) {
      const float hzenv = accz[t][v] * g.w;
      const float cI = (accr[t][v]     + bias[t])     * hzenv;
      const float cA = (accr[2 + t][v] + bias[2 + t]) * hzenv;
      const float cS = (accr[4 + t][v] + bias[4 + t]) * hzenv;
      float* pI = outI + rowbase + t * 144;
      pI[0] = cI;   pI[1] = 0.0f; pI[2] = 0.0f;
      pI[3] = 0.0f; pI[4] = cI;   pI[5] = 0.0f;
      pI[6] = 0.0f; pI[7] = 0.0f; pI[8] = cI;
      float* pA = outA + rowbase + t * 144;
      pA[0] = 0.0f;       pA[1] = -g.z * cA; pA[2] =  g.y * cA;
      pA[3] =  g.z * cA;  pA[4] = 0.0f;      pA[5] = -g.x * cA;
      pA[6] = -g.y * cA;  pA[7] =  g.x * cA; pA[8] = 0.0f;
      float* pS = outS + rowbase + t * 144;
      pS[0] = s00 * cS; pS[1] = s01 * cS; pS[2] = s02 * cS;
      pS[3] = s01 * cS; pS[4] = s11 * cS; pS[5] = s12 * cS;
      pS[6] = s02 * cS; pS[7] = s12 * cS; pS[8] = s22 * cS;
    }
  }
}

extern "C" void kernel_launch(void* const* d_in, const int* in_sizes, int n_in,
                              void* d_out, int out_size, void* d_ws, size_t ws_size,
                              hipStream_t stream) {
  (void)n_in; (void)out_size; (void)d_ws; (void)ws_size;
  const int*   an  = (const int*)  d_in[0];
  const int*   nbr = (const int*)  d_in[1];
  const float* ev  = (const float*)d_in[2];
  const float* zt  = (const float*)d_in[3];
  const float* zw  = (const float*)d_in[4];
  const float* dw  = (const float*)d_in[5];
  const float* db  = (const float*)d_in[6];
  const float* rb  = (const float*)d_in[7];
  const float* rc  = (const float*)d_in[8];
  float* out = (float*)d_out;

  const int nEdges = in_sizes[2] / 3;       // 160000
  const int nTiles = (nEdges + 15) / 16;    // 10000
  const int wavesPerBlock = 256 / 32;
  const int blocks = (nTiles + wavesPerBlock - 1) / wavesPerBlock;
  edge_embed_kernel<<<blocks, 256, 0, stream>>>(
      an, nbr, ev, zt, zw, dw, db, rb, rc, out, nEdges, nTiles);
}